// FINSystem_42279658062194
// MI455X (gfx1250) — compile-verified
//
#include <hip/hip_runtime.h>
#include <hip/hip_bf16.h>
#include <math.h>

typedef __attribute__((ext_vector_type(2))) float v2f;
typedef __attribute__((ext_vector_type(8))) float v8f;

#define NN   131072
#define KKN  16
#define HDCC 1024
#define NBLK 512
#define HH   64
#define DTT  0.05f
#define STEPS 4

// ---- workspace layout (float offsets) ----
constexpr int WS_W      = 0;                       // N*K
constexpr int WS_PHI0   = WS_W      + NN * KKN;
constexpr int WS_PHI1   = WS_PHI0   + NN;
constexpr int WS_AMP0   = WS_PHI1   + NN;
constexpr int WS_AMP1   = WS_AMP0   + NN;
constexpr int WS_HVSIN  = WS_AMP1   + NN;          // HDC
constexpr int WS_H      = WS_HVSIN  + HDCC;        // 64
constexpr int WS_STATS  = WS_H      + HH;          // 8 (omega mean/std)
constexpr int WS_YREG   = WS_STATS  + 8;           // 8 (y[6], reg)
constexpr int WS_USUM   = WS_YREG   + 8;           // NBLK*4
constexpr int WS_OMPART = WS_USUM   + NBLK * 4;    // NBLK*2
constexpr int WS_DPART  = WS_OMPART + NBLK * 2;    // NBLK
constexpr int WS_HVPART = WS_DPART  + NBLK;        // NBLK*HDC

// output layout
constexpr int OUT_Y   = 4;
constexpr int OUT_HV  = 4 + STEPS * 6;             // 28
constexpr int OUT_ACT = OUT_HV + STEPS * HDCC;     // 4124

__device__ __forceinline__ float softplusf(float x) {
    return fmaxf(x, 0.0f) + log1pf(__expf(-fabsf(x)));
}

// branch-free fast tanh: 1 - 2/(exp(2x)+1), clamp avoids exp overflow
__device__ __forceinline__ float tanh_fast(float x) {
    float xc = fminf(fmaxf(x, -15.0f), 15.0f);
    float e  = __expf(2.0f * xc);
    return 1.0f - __fdividef(2.0f, e + 1.0f);
}

// ---------------- precompute: w, initial buffers ----------------
__global__ void k_prep(const float* __restrict__ phi, const float* __restrict__ amp,
                       const float* __restrict__ pos, const int* __restrict__ nbr,
                       float* __restrict__ ws) {
    int n = blockIdx.x * 256 + threadIdx.x;
    float px = pos[n * 3 + 0], py = pos[n * 3 + 1], pz = pos[n * 3 + 2];
#pragma unroll
    for (int k = 0; k < KKN; ++k) {
        int j = nbr[n * KKN + k];
        float dx = pos[j * 3 + 0] - px;
        float dy = pos[j * 3 + 1] - py;
        float dz = pos[j * 3 + 2] - pz;
        ws[WS_W + n * KKN + k] = __expf(-(dx * dx + dy * dy + dz * dz));
    }
    ws[WS_PHI0 + n] = phi[n];
    ws[WS_AMP0 + n] = amp[n];
}

// ---------------- precompute: hv_sin partials + omega stats partials ----------------
__global__ void k_hvsin_part(const float* __restrict__ omega,
                             const float* __restrict__ basis_sin,
                             float* __restrict__ ws) {
    __shared__ __align__(16) float sv[256];
    __shared__ float om[256];
    const int t = threadIdx.x, b = blockIdx.x;
    const int n = b * 256 + t;
    float o = omega[n];
    om[t] = o;
    sv[t] = __sinf(o);
    __syncthreads();
    float ax = 0.f, ay = 0.f, az = 0.f, aw = 0.f;
    const float* basep = basis_sin + (size_t)b * 256 * HDCC + 4 * t;
    for (int r0 = 0; r0 < 256; r0 += 16) {
        if (b * 256 + r0 + 32 < NN)
            __builtin_prefetch(basep + (size_t)(r0 + 32) * HDCC, 0, 1);
#pragma unroll
        for (int q = 0; q < 4; ++q) {
            float4 cv = *(const float4*)&sv[r0 + 4 * q];
            float cc[4] = {cv.x, cv.y, cv.z, cv.w};
#pragma unroll
            for (int rr = 0; rr < 4; ++rr) {
                int r = r0 + 4 * q + rr;
                float4 v = *(const float4*)(basep + (size_t)r * HDCC);
                float c = cc[rr];
                ax = fmaf(c, v.x, ax); ay = fmaf(c, v.y, ay);
                az = fmaf(c, v.z, az); aw = fmaf(c, v.w, aw);
            }
        }
    }
    float4 res; res.x = ax; res.y = ay; res.z = az; res.w = aw;
    *(float4*)(ws + WS_HVPART + b * HDCC + 4 * t) = res;
    __syncthreads();
    if (t == 0) {
        float s = 0.f, s2 = 0.f;
        for (int i = 0; i < 256; ++i) { s += om[i]; s2 += om[i] * om[i]; }
        ws[WS_OMPART + b * 2 + 0] = s;
        ws[WS_OMPART + b * 2 + 1] = s2;
    }
}

__global__ void k_hvsin_reduce(float* __restrict__ ws) {
    const int b = blockIdx.x, t = threadIdx.x;
    if (b < 4) {
        int d = b * 256 + t;
        float s = 0.f;
        for (int i = 0; i < NBLK; ++i) s += ws[WS_HVPART + i * HDCC + d];
        ws[WS_HVSIN + d] = s;
    } else if (t == 0) {
        float s = 0.f, s2 = 0.f;
        for (int i = 0; i < NBLK; ++i) { s += ws[WS_OMPART + i * 2]; s2 += ws[WS_OMPART + i * 2 + 1]; }
        float m = s / (float)NN;
        ws[WS_STATS + 0] = m;
        ws[WS_STATS + 1] = sqrtf(fmaxf(s2 / (float)NN - m * m, 0.f));
    }
}

// ---------------- step 1: coupling, phi update, u partial sums ----------------
__global__ void k_phase(const float* __restrict__ omega, const int* __restrict__ nbr,
                        const float* __restrict__ cscale,
                        const float* __restrict__ phiC, float* __restrict__ phiN,
                        const float* __restrict__ ampC, float* __restrict__ ws) {
    __shared__ float4 rs[256];
    const int t = threadIdx.x, b = blockIdx.x;
    const int n = b * 256 + t;
    float phi = phiC[n], amp = ampC[n];
    float c = 0.f;
#pragma unroll
    for (int k = 0; k < KKN; ++k) {
        int j = nbr[n * KKN + k];
        c += ws[WS_W + n * KKN + k] * __sinf(phiC[j] - phi);
    }
    float coupling = cscale[0] * c * (1.0f / (float)KKN);
    phiN[n] = phi + DTT * (omega[n] + coupling);
    float sp, cp;
    __sincosf(phi, &sp, &cp);
    float4 v; v.x = sp; v.y = cp; v.z = amp; v.w = amp * amp;
    rs[t] = v;
    __syncthreads();
    for (int off = 128; off > 0; off >>= 1) {
        if (t < off) {
            rs[t].x += rs[t + off].x; rs[t].y += rs[t + off].y;
            rs[t].z += rs[t + off].z; rs[t].w += rs[t + off].w;
        }
        __syncthreads();
    }
    if (t == 0) *(float4*)(ws + WS_USUM + b * 4) = rs[0];
}

// ---------------- step 2: SSM update (1 block) ----------------
__global__ void k_ssm(const float* __restrict__ A_raw, const float* __restrict__ Bm,
                      const float* __restrict__ Cmat, const float* __restrict__ h0,
                      float* __restrict__ ws, float* __restrict__ out, int s) {
    __shared__ float su[4];
    __shared__ float u[6];
    __shared__ float hl[HH];
    __shared__ float yl[6];
    const int t = threadIdx.x;
    if (t < 4) {
        float a = 0.f;
        for (int b = 0; b < NBLK; ++b) a += ws[WS_USUM + b * 4 + t];
        su[t] = a;
    }
    __syncthreads();
    if (t == 0) {
        u[0] = ws[WS_STATS + 0];
        u[1] = ws[WS_STATS + 1];
        u[2] = su[0] * (1.f / (float)NN);
        u[3] = su[1] * (1.f / (float)NN);
        float m = su[2] * (1.f / (float)NN);
        u[4] = m;
        u[5] = sqrtf(fmaxf(su[3] * (1.f / (float)NN) - m * m, 0.f));
    }
    __syncthreads();
    if (t < HH) {
        float a = softplusf(A_raw[t]);
        float dec = __expf(-a * DTT);
        float hp = (s == 0) ? h0[t] : ws[WS_H + t];
        float bu = 0.f;
#pragma unroll
        for (int j = 0; j < 6; ++j) bu += Bm[t * 6 + j] * u[j];
        float h = hp * dec + DTT * bu;
        ws[WS_H + t] = h;
        hl[t] = h;
    }
    __syncthreads();
    if (t < 6) {
        float y = 0.f;
        for (int i = 0; i < HH; ++i) y += Cmat[t * HH + i] * hl[i];
        yl[t] = y;
        ws[WS_YREG + t] = y;
        out[OUT_Y + s * 6 + t] = y;
    }
    __syncthreads();
    if (t == 0) {
        float r = 0.f;
        for (int j = 0; j < 6; ++j) r += yl[j] * yl[j] + u[j] * u[j];
        ws[WS_YREG + 6] = 0.01f * r;
    }
}

// ---------------- step 3: amp/mod update, WMMA MLP actions, hv matvec partials ----------------
__global__ void __launch_bounds__(256)
k_field(const float* __restrict__ omega, const float* __restrict__ Wp,
        const float* __restrict__ bp, const float* __restrict__ basis_cos,
        const float* __restrict__ W1, const float* __restrict__ b1,
        const float* __restrict__ W2, const float* __restrict__ b2,
        const float* __restrict__ adr, const float* __restrict__ mgr,
        const float* __restrict__ phiD, const float* __restrict__ ampC,
        float* __restrict__ ampN, float* __restrict__ ws,
        float* __restrict__ out, int s) {
    __shared__ float4 feat_s[256];                  // [omega, sin, cos, amp]
    __shared__ __align__(16) float cph_s[256];      // cos(phi) for matvec
    __shared__ float W2p[64][16];                   // zero-padded W2 (64x3 -> 64x16)
    __shared__ float hid[8][16][68];                // per-wave hidden tile (A-layout src)
    const int t = threadIdx.x, b = blockIdx.x;
    const int n = b * 256 + t;

    for (int idx = t; idx < 64 * 16; idx += 256) {
        int r = idx >> 4, c0 = idx & 15;
        W2p[r][c0] = (c0 < 3) ? W2[r * 3 + c0] : 0.0f;
    }

    // per-element scalar work
    float yv[6];
#pragma unroll
    for (int j = 0; j < 6; ++j) yv[j] = ws[WS_YREG + j];
    float ad = softplusf(adr[0]);
    float mg = softplusf(mgr[0]);
    float dotp = bp[n];
#pragma unroll
    for (int j = 0; j < 6; ++j) dotp += Wp[n * 6 + j] * yv[j];
    float mod = tanh_fast(dotp);
    float ac = ampC[n];
    float an = fmaxf(ac + DTT * (-ad * ac + mg * mod), 0.001f);
    ampN[n] = an;
    float ph = phiD[n];
    float sph, cph;
    __sincosf(ph, &sph, &cph);
    float4 f4; f4.x = omega[n]; f4.y = sph; f4.z = cph; f4.w = an;
    feat_s[t] = f4;
    cph_s[t] = cph;
    __syncthreads();

    // ---- actions MLP via V_WMMA_F32_16X16X4_F32 ----
    const int lane = t & 31;
    const int w    = t >> 5;
    const int lr16 = lane & 15;
    const int hi   = lane >> 4;     // 0: K=0,1   1: K=2,3

    v2f b1t[4];
#pragma unroll
    for (int tt = 0; tt < 4; ++tt) {
        int col = tt * 16 + lr16;
        b1t[tt].x = W1[(hi * 2 + 0) * 64 + col];
        b1t[tt].y = W1[(hi * 2 + 1) * 64 + col];
    }
#pragma unroll
    for (int g = 0; g < 2; ++g) {
        int lrow = w * 32 + g * 16 + lr16;      // local row in block
        float4 fr = feat_s[lrow];
        v2f a;
        a.x = hi ? fr.z : fr.x;
        a.y = hi ? fr.w : fr.y;
        // layer 1: D[16x64] in 4 tiles
        v8f c0, c1, c2t, c3;
        {
            v8f z = {};
            c0  = __builtin_amdgcn_wmma_f32_16x16x4_f32(false, a, false, b1t[0], (short)0, z, false, false);
            c1  = __builtin_amdgcn_wmma_f32_16x16x4_f32(false, a, false, b1t[1], (short)0, z, false, false);
            c2t = __builtin_amdgcn_wmma_f32_16x16x4_f32(false, a, false, b1t[2], (short)0, z, false, false);
            c3  = __builtin_amdgcn_wmma_f32_16x16x4_f32(false, a, false, b1t[3], (short)0, z, false, false);
        }
        // bias + tanh -> LDS in A-operand layout
        v8f ctile[4] = {c0, c1, c2t, c3};
#pragma unroll
        for (int tt = 0; tt < 4; ++tt) {
            int col = tt * 16 + lr16;
            float bias = b1[col];
#pragma unroll
            for (int r = 0; r < 8; ++r) {
                int m = r + 8 * hi;
                hid[w][m][col] = tanh_fast(ctile[tt][r] + bias);
            }
        }
        __syncthreads();
        // layer 2: K=64 via 16 chained accumulate WMMAs
        v8f c2acc = {};
#pragma unroll
        for (int k = 0; k < 16; ++k) {
            int kk = 4 * k + 2 * hi;
            v2f a2, b2t;
            a2.x  = hid[w][lr16][kk];
            a2.y  = hid[w][lr16][kk + 1];
            b2t.x = W2p[kk][lr16];
            b2t.y = W2p[kk + 1][lr16];
            c2acc = __builtin_amdgcn_wmma_f32_16x16x4_f32(false, a2, false, b2t, (short)0, c2acc, false, false);
        }
        if (lr16 < 3) {
            float bb = b2[lr16];
            int rowbase = b * 256 + w * 32 + g * 16;
#pragma unroll
            for (int r = 0; r < 8; ++r) {
                int m = r + 8 * hi;
                out[OUT_ACT + (size_t)s * NN * 3 + (size_t)(rowbase + m) * 3 + lr16] =
                    tanh_fast(c2acc[r] + bb);
            }
        }
        __syncthreads();
    }

    // ---- hv matvec partials: acc += cos(phi_d[n]) * basis_cos[n, :] ----
    float ax = 0.f, ay = 0.f, az = 0.f, aw = 0.f;
    const float* basep = basis_cos + (size_t)b * 256 * HDCC + 4 * t;
    for (int r0 = 0; r0 < 256; r0 += 16) {
        if (b * 256 + r0 + 32 < NN)
            __builtin_prefetch(basep + (size_t)(r0 + 32) * HDCC, 0, 1);
#pragma unroll
        for (int q = 0; q < 4; ++q) {
            float4 cv = *(const float4*)&cph_s[r0 + 4 * q];
            float cc[4] = {cv.x, cv.y, cv.z, cv.w};
#pragma unroll
            for (int rr = 0; rr < 4; ++rr) {
                int r = r0 + 4 * q + rr;
                float4 v = *(const float4*)(basep + (size_t)r * HDCC);
                float c = cc[rr];
                ax = fmaf(c, v.x, ax); ay = fmaf(c, v.y, ay);
                az = fmaf(c, v.z, az); aw = fmaf(c, v.w, aw);
            }
        }
    }
    float4 res; res.x = ax; res.y = ay; res.z = az; res.w = aw;
    *(float4*)(ws + WS_HVPART + b * HDCC + 4 * t) = res;
}

// ---------------- step 4: dissonance edge partials ----------------
__global__ void k_diss(const int* __restrict__ nbr, const float* __restrict__ phiD,
                       const float* __restrict__ ampD, float* __restrict__ ws) {
    __shared__ float red[256];
    const int t = threadIdx.x, b = blockIdx.x;
    const int n = b * 256 + t;
    float pi = phiD[n], ai = ampD[n];
    float e = 0.f;
#pragma unroll
    for (int k = 0; k < KKN; ++k) {
        int j = nbr[n * KKN + k];
        e += (1.0f - __cosf(phiD[j] - pi)) * ai * ampD[j] * ws[WS_W + n * KKN + k];
    }
    red[t] = e;
    __syncthreads();
    for (int off = 128; off > 0; off >>= 1) {
        if (t < off) red[t] += red[t + off];
        __syncthreads();
    }
    if (t == 0) ws[WS_DPART + b] = red[0];
}

// ---------------- step 5: finalize hv sign + diss ----------------
__global__ void k_final(float* __restrict__ ws, float* __restrict__ out, int s) {
    const int b = blockIdx.x, t = threadIdx.x;
    if (b < 4) {
        int d = b * 256 + t;
        float ssum = ws[WS_HVSIN + d];
        for (int i = 0; i < NBLK; ++i) ssum += ws[WS_HVPART + i * HDCC + d];
        out[OUT_HV + s * HDCC + d] = (ssum > 0.f) ? 1.f : ((ssum < 0.f) ? -1.f : 0.f);
    } else {
        __shared__ float red[256];
        float v = ws[WS_DPART + t] + ws[WS_DPART + 256 + t];
        red[t] = v;
        __syncthreads();
        for (int off = 128; off > 0; off >>= 1) {
            if (t < off) red[t] += red[t + off];
            __syncthreads();
        }
        if (t == 0)
            out[s] = red[0] * (1.0f / ((float)NN * (float)KKN)) + ws[WS_YREG + 6];
    }
}

extern "C" void kernel_launch(void* const* d_in, const int* in_sizes, int n_in,
                              void* d_out, int out_size, void* d_ws, size_t ws_size,
                              hipStream_t stream) {
    (void)in_sizes; (void)n_in; (void)out_size; (void)ws_size;
    const float* phi       = (const float*)d_in[0];
    const float* omega     = (const float*)d_in[1];
    const float* amplitude = (const float*)d_in[2];
    const float* position  = (const float*)d_in[3];
    const float* A_raw     = (const float*)d_in[4];
    const float* Bm        = (const float*)d_in[5];
    const float* Cm        = (const float*)d_in[6];
    const float* Wp        = (const float*)d_in[7];
    const float* bp        = (const float*)d_in[8];
    const float* basis_cos = (const float*)d_in[9];
    const float* basis_sin = (const float*)d_in[10];
    const float* W1        = (const float*)d_in[11];
    const float* b1        = (const float*)d_in[12];
    const float* W2        = (const float*)d_in[13];
    const float* b2        = (const float*)d_in[14];
    const float* adr       = (const float*)d_in[15];
    const float* mgr       = (const float*)d_in[16];
    const float* cscale    = (const float*)d_in[17];
    const float* h0        = (const float*)d_in[18];
    const int*   nbr       = (const int*)d_in[19];
    float* out = (float*)d_out;
    float* ws  = (float*)d_ws;

    k_prep<<<NBLK, 256, 0, stream>>>(phi, amplitude, position, nbr, ws);
    k_hvsin_part<<<NBLK, 256, 0, stream>>>(omega, basis_sin, ws);
    k_hvsin_reduce<<<5, 256, 0, stream>>>(ws);

    for (int s = 0; s < STEPS; ++s) {
        float* pc = ws + ((s & 1) ? WS_PHI1 : WS_PHI0);
        float* pn = ws + ((s & 1) ? WS_PHI0 : WS_PHI1);
        float* ac = ws + ((s & 1) ? WS_AMP1 : WS_AMP0);
        float* an = ws + ((s & 1) ? WS_AMP0 : WS_AMP1);
        k_phase<<<NBLK, 256, 0, stream>>>(omega, nbr, cscale, pc, pn, ac, ws);
        k_ssm<<<1, 256, 0, stream>>>(A_raw, Bm, Cm, h0, ws, out, s);
        k_field<<<NBLK, 256, 0, stream>>>(omega, Wp, bp, basis_cos, W1, b1, W2, b2,
                                          adr, mgr, pn, ac, an, ws, out, s);
        k_diss<<<NBLK, 256, 0, stream>>>(nbr, pn, an, ws);
        k_final<<<5, 256, 0, stream>>>(ws, out, s);
    }
}